// CoTGlow_33071248179506
// MI455X (gfx1250) — compile-verified
//
#include <hip/hip_runtime.h>
#include <math.h>

// ---------------------------------------------------------------------------
// CoTGlow log-prob on MI455X (gfx1250, wave32, WMMA f32 16x16x4 + TDM).
// Frequency-domain restructure: no upsampled FFT, no inverse FFT.
//   Bz(c, b*1024+k*32+l)      = Z_lvl(c)(k mod h, l mod w)       (analytic DFT)
//   GEMM (per m):  V(d,col)   = sum_c LaT(m,d,c) * Bz(c,col)     (WMMA f32,
//                                A/B tiles staged to LDS by tensor_load_to_lds)
//   epilogue:      *= i*lam_a ; alias-fold (1/r^2) ; * 1/sqrt(hw) -> Vdown
//   S_zz = Y Y^T, S_zw = Y.Wf (WMMA f32, K=24576), then 32x32 Cholesky.
// ---------------------------------------------------------------------------

typedef float v2f __attribute__((ext_vector_type(2)));
typedef float v8f __attribute__((ext_vector_type(8)));
typedef unsigned int u32x4 __attribute__((ext_vector_type(4)));
typedef int i32x4 __attribute__((ext_vector_type(4)));
typedef int i32x8 __attribute__((ext_vector_type(8)));

#define CT    282
#define CTP   288
#define MBASE 32
#define BATCH 4
#define NCOL  4096
#define TPB   12288          // complex elements per (b,m) in Vdown / per b in Wf
#define KB    32             // k-panel depth for the big GEMM
#define NPAN  (CTP / KB)     // 9 panels
#define PI_F  3.14159265358979323846f

// ---- workspace layout (float offsets) ----
#define SZ_B     (CTP*NCOL)                  // 1,179,648
#define OFF_BRE  0u
#define OFF_BIM  (OFF_BRE + SZ_B)
#define OFF_VD   (OFF_BIM + SZ_B)
#define SZ_VD    (2u*BATCH*MBASE*TPB)        // 3,145,728
#define OFF_SWW  (OFF_VD + SZ_VD)
#define OFF_LAT  (OFF_SWW + 4u)
#define SZ_LAT   (MBASE*CTP*CTP)             // 2,654,208
#define OFF_LAM  (OFF_LAT + SZ_LAT)
#define OFF_WF   (OFF_LAM + MBASE*1024u)
#define SZ_WF    (2u*BATCH*TPB)              // 98,304
#define OFF_SZZ  (OFF_WF + SZ_WF)
#define OFF_SZW  (OFF_SZZ + BATCH*MBASE*MBASE)
#define ZERO_CNT OFF_LAT                     // zero: Bre,Bim,Vdown,Sww

__constant__ int   LV_C[5]  = {192, 48, 24, 12, 6};
__constant__ int   LV_H[5]  = {2, 4, 8, 16, 32};
__constant__ int   LV_CO[5] = {0, 192, 240, 264, 276};
__constant__ int   LV_O2[5] = {0, 768, 1536, 3072, 6144};
__constant__ float LV_S[5]  = {1.f/512.f, 1.f/256.f, 1.f/128.f, 1.f/64.f, 1.f/32.f};

__device__ __forceinline__ int lvl_of(int c) {
  return c < 192 ? 0 : (c < 240 ? 1 : (c < 264 ? 2 : (c < 276 ? 3 : 4)));
}

// ---------------------------------------------------------------------------
// Tensor Data Mover: 2D tile (tile0 contiguous elems x tile1 lines) of f32
// from global to LDS.  D# packing per CDNA5 ISA ch.8 (group0: count/lds/gaddr/
// type=2; group1: data_size=4B, dims, strides, optional LDS padding).
__device__ __forceinline__ void tdm_load_2d(unsigned lds_off, const void* gptr,
                                            int tile0, int tile1,
                                            long long stride0_elems,
                                            unsigned padctl) {
  unsigned long long ga = (unsigned long long)(uintptr_t)gptr;
  u32x4 g0;
  g0[0] = 1u;                                               // count=1 (valid)
  g0[1] = lds_off;                                          // lds_addr (bytes)
  g0[2] = (unsigned)ga;                                     // global addr lo32
  g0[3] = (unsigned)((ga >> 32) & 0x1FFFFFFull) | (2u << 30);  // hi25 | type=2
  i32x8 g1;
  g1[0] = (int)(0x00020000u | padctl);        // wg_mask=0, data_size=4B, pad
  g1[1] = 0;                                  // abar=0, tensor_dim0 lo16 = 0
  g1[2] = 0x4000;                             // tensor_dim0 hi16 (=2^30), td1 lo16=0
  g1[3] = (int)(0x4000u | ((unsigned)tile0 << 16));  // td1 hi16 (=2^30), tile_dim0
  g1[4] = tile1;                              // tile_dim1, tile_dim2=0
  g1[5] = (int)(stride0_elems & 0xFFFFFFFFll);       // dim0 stride lo32
  g1[6] = (int)((stride0_elems >> 32) & 0xFFFFll);   // dim0 stride hi16, d1s lo=0
  g1[7] = 0;                                  // dim1 stride hi
  i32x4 gz = {0, 0, 0, 0};
#if defined(__clang_major__) && (__clang_major__ >= 23)
  i32x8 gz8 = {0, 0, 0, 0, 0, 0, 0, 0};
  __builtin_amdgcn_tensor_load_to_lds(g0, g1, gz, gz, gz8, 0);
#else
  __builtin_amdgcn_tensor_load_to_lds(g0, g1, gz, gz, 0);
#endif
}

// ---------------------------------------------------------------------------
__global__ void k_zero(float* p, unsigned n) {
  for (unsigned i = blockIdx.x * blockDim.x + threadIdx.x; i < n;
       i += gridDim.x * blockDim.x)
    p[i] = 0.f;
}

// lam_a imaginary part: lam(k,l) - lam((-k)%32, (-l)%32)
__global__ void k_lam(const float* lam, float* lamim) {
  int i = blockIdx.x * blockDim.x + threadIdx.x;
  if (i >= MBASE * 1024) return;
  int m = i >> 10, k = (i >> 5) & 31, l = i & 31;
  lamim[i] = lam[i] - lam[m * 1024 + ((32 - k) & 31) * 32 + ((32 - l) & 31)];
}

// A(m, d, c) = La(m, c, d) = L[m][c][d] - L[m][d][c], zero-padded to 288x288
__global__ void k_lat(const float* L, float* A) {
  int i = blockIdx.x * blockDim.x + threadIdx.x;
  if (i >= MBASE * CTP * CTP) return;
  int m = i / (CTP * CTP);
  int r = (i / CTP) % CTP;
  int c = i % CTP;
  float v = 0.f;
  if (r < CT && c < CT)
    v = L[(m * CT + c) * CT + r] - L[(m * CT + r) * CT + c];
  A[i] = v;
}

// Bz(c, col=(b,k,l)) = DFT_lvl(z)(b, dd, k mod h, l mod w)  (analytic tiling)
__global__ void k_zfft(const float* z0, const float* z1, const float* z2,
                       const float* z3, const float* z4,
                       float* Bre, float* Bim) {
  int i = blockIdx.x * blockDim.x + threadIdx.x;
  if (i >= CT * NCOL) return;
  int c = i / NCOL, col = i % NCOL;
  int b = col >> 10, k = (col >> 5) & 31, l = col & 31;
  int j = lvl_of(c);
  int h = LV_H[j];
  int dd = c - LV_CO[j];
  const float* z = (j == 0) ? z0 : (j == 1) ? z1 : (j == 2) ? z2 : (j == 3) ? z3 : z4;
  int kk = k & (h - 1), ll = l & (h - 1);
  const float* zp = z + (size_t)(b * LV_C[j] + dd) * h * h;
  float wk = -2.f * PI_F * (float)kk / (float)h;
  float wl = -2.f * PI_F * (float)ll / (float)h;
  float re = 0.f, im = 0.f;
  for (int n = 0; n < h; n++) {
    float an = wk * (float)n;
    for (int mm = 0; mm < h; mm++) {
      float ang = an + wl * (float)mm;
      float zv = zp[n * h + mm];
      re += zv * cosf(ang);
      im += zv * sinf(ang);
    }
  }
  Bre[i] = re;
  Bim[i] = im;
}

// ---------------------------------------------------------------------------
// Big batched GEMM: per m, out(d,col) = sum_c A(d,c)*Bz(c,col), complex B.
// Double-buffered TDM staging: per 32-deep k-panel, wave 0 issues three
// tensor_load_to_lds (A 96x32 with LDS pad->stride 33; Bre/Bim 32x64),
// waits TENSORcnt, workgroup barrier, all 6 waves run WMMA out of LDS.
// Epilogue multiplies by i*lam_a, alias-folds into Vdown with atomics.
// Grid (64, 3, 32), block 192 (6 waves); wave tile 16x64, 4 col-subtiles x re/im.
#define PAD_A 0x01100000u  // pad_enable | pad_interval=32dw (code4) | amount=1dw

__global__ __launch_bounds__(192) void k_gemm(const float* __restrict__ A_,
                                              const float* __restrict__ Bre,
                                              const float* __restrict__ Bim,
                                              const float* __restrict__ lamim,
                                              float* __restrict__ Vd) {
  __shared__ float Alds[2][96][33];   // TDM pads each 32-float line by 1
  __shared__ float Brl[2][KB][64];
  __shared__ float Bil[2][KB][64];

  const int m = blockIdx.z;
  const int lane = threadIdx.x & 31, wv = threadIdx.x >> 5;
  const int rowblk = blockIdx.y * 96;
  const int colbase = blockIdx.x * 64;
  const int half = lane >> 4, lr = lane & 15;
  const float* Abase = A_ + (size_t)m * CTP * CTP + (size_t)rowblk * CTP;

  v8f zero = {0.f, 0.f, 0.f, 0.f, 0.f, 0.f, 0.f, 0.f};
  v8f accR[4], accI[4];
#pragma unroll
  for (int t = 0; t < 4; t++) { accR[t] = zero; accI[t] = zero; }

  auto issue = [&](int p, int buf) {
    tdm_load_2d((unsigned)(uintptr_t)&Alds[buf][0][0], Abase + p * KB,
                KB, 96, (long long)CTP, PAD_A);
    tdm_load_2d((unsigned)(uintptr_t)&Brl[buf][0][0],
                Bre + (size_t)p * KB * NCOL + colbase, 64, KB,
                (long long)NCOL, 0u);
    tdm_load_2d((unsigned)(uintptr_t)&Bil[buf][0][0],
                Bim + (size_t)p * KB * NCOL + colbase, 64, KB,
                (long long)NCOL, 0u);
  };

  if (wv == 0) { issue(0, 0); issue(1, 1); }

  for (int p = 0; p < NPAN; ++p) {
    if (wv == 0) {
      if (p + 1 < NPAN) __builtin_amdgcn_s_wait_tensorcnt(3);
      else              __builtin_amdgcn_s_wait_tensorcnt(0);
    }
    __syncthreads();
    const int buf = p & 1;
    const int arow = wv * 16 + lr;
    for (int kk = 0; kk < KB; kk += 4) {
      int kb = kk + 2 * half;
      v2f a = {Alds[buf][arow][kb], Alds[buf][arow][kb + 1]};
#pragma unroll
      for (int t = 0; t < 4; t++) {
        int cc = 16 * t + lr;
        v2f bre = {Brl[buf][kb][cc], Brl[buf][kb + 1][cc]};
        v2f bim = {Bil[buf][kb][cc], Bil[buf][kb + 1][cc]};
        accR[t] = __builtin_amdgcn_wmma_f32_16x16x4_f32(
            false, a, false, bre, (short)0, accR[t], false, false);
        accI[t] = __builtin_amdgcn_wmma_f32_16x16x4_f32(
            false, a, false, bim, (short)0, accI[t], false, false);
      }
    }
    __syncthreads();
    if (wv == 0 && p + 2 < NPAN) issue(p + 2, buf);
  }

  // epilogue: v = (i*lam)*(re + i*im) => re' = -lam*im, im' = lam*re
  const int rowbase = rowblk + wv * 16;
#pragma unroll
  for (int t = 0; t < 4; t++) {
    int col = colbase + 16 * t + lr;
    int b = col >> 10, k = (col >> 5) & 31, l = col & 31;
    float lam = lamim[m * 1024 + k * 32 + l];
#pragma unroll
    for (int i2 = 0; i2 < 8; i2++) {
      int row = rowbase + half * 8 + i2;
      if (row < CT) {
        int j = lvl_of(row);
        int h = LV_H[j];
        int dd = row - LV_CO[j];
        float s = LV_S[j];  // 1/(r^2) * 1/sqrt(h*w)
        int idx = (b * MBASE + m) * TPB + LV_O2[j] +
                  (dd * h + (k & (h - 1))) * h + (l & (h - 1));
        atomicAdd(&Vd[2 * idx],     -lam * accI[t][i2] * s);
        atomicAdd(&Vd[2 * idx + 1],  lam * accR[t][i2] * s);
      }
    }
  }
}

// Wf(b, t) = DFT(w_lvl) / sqrt(h*w), same flat (level,dd,k,l) ordering as Vdown
__global__ void k_wfft(const float* w0, const float* w1, const float* w2,
                       const float* w3, const float* w4, float* Wf) {
  int i = blockIdx.x * blockDim.x + threadIdx.x;
  if (i >= BATCH * TPB) return;
  int b = i / TPB, t = i % TPB;
  int j = (t < 768) ? 0 : (t < 1536) ? 1 : (t < 3072) ? 2 : (t < 6144) ? 3 : 4;
  int h = LV_H[j];
  int hw = h * h;
  int rem = t - LV_O2[j];
  int dd = rem / hw, pos = rem % hw, kk = pos / h, ll = pos % h;
  const float* w = (j == 0) ? w0 : (j == 1) ? w1 : (j == 2) ? w2 : (j == 3) ? w3 : w4;
  const float* wp = w + (size_t)(b * LV_C[j] + dd) * hw;
  float wk = -2.f * PI_F * (float)kk / (float)h;
  float wl = -2.f * PI_F * (float)ll / (float)h;
  float re = 0.f, im = 0.f;
  for (int n = 0; n < h; n++) {
    float an = wk * (float)n;
    for (int mm = 0; mm < h; mm++) {
      float ang = an + wl * (float)mm;
      float wv = wp[n * h + mm];
      re += wv * cosf(ang);
      im += wv * sinf(ang);
    }
  }
  float sc = 1.f / (float)h;  // 1/sqrt(h*w)
  Wf[2 * i] = re * sc;
  Wf[2 * i + 1] = im * sc;
}

// Gram GEMM: per b, S_zz = Y Y^T (32x32), S_zw = Y . Wf ; K = 24576 (WMMA f32)
// Block = 192 threads (6 waves): wave -> (mt in {0,1}, nt in {0,1,2}); N padded to 48.
__global__ __launch_bounds__(192) void k_gram(const float* __restrict__ Vd,
                                              const float* __restrict__ Wf,
                                              float* __restrict__ Szz,
                                              float* __restrict__ Szw) {
  int b = blockIdx.x;
  int lane = threadIdx.x & 31, wv = threadIdx.x >> 5;
  int mt = wv & 1, nt = wv >> 1;
  int half = lane >> 4, lr = lane & 15;
  const float* Y = Vd + (size_t)b * MBASE * 2 * TPB;  // Y[n][k], k in [0,24576)
  v8f acc = {0.f, 0.f, 0.f, 0.f, 0.f, 0.f, 0.f, 0.f};
  int arow = mt * 16 + lr;
  int col = nt * 16 + lr;
  const float* Ap = Y + (size_t)arow * (2 * TPB);
  for (int kk = 0; kk < 2 * TPB; kk += 4) {
    int kb = kk + 2 * half;
    v2f a = {Ap[kb], Ap[kb + 1]};
    float b0 = 0.f, b1 = 0.f;
    if (col < 32) {
      const float* Bp = Y + (size_t)col * (2 * TPB);
      b0 = Bp[kb]; b1 = Bp[kb + 1];
    } else if (col == 32) {
      b0 = Wf[b * 2 * TPB + kb]; b1 = Wf[b * 2 * TPB + kb + 1];
    }
    v2f bb = {b0, b1};
    acc = __builtin_amdgcn_wmma_f32_16x16x4_f32(
        false, a, false, bb, (short)0, acc, false, false);
  }
#pragma unroll
  for (int i2 = 0; i2 < 8; i2++) {
    int row = mt * 16 + half * 8 + i2;
    if (col < 32)       Szz[b * 1024 + row * 32 + col] = acc[i2];
    else if (col == 32) Szw[b * 32 + row] = acc[i2];
  }
}

__global__ void k_ww(const float* w, int n, int perB, float* sww) {
  int i = blockIdx.x * blockDim.x + threadIdx.x;
  if (i < n) atomicAdd(&sww[i / perB], w[i] * w[i]);
}

// Final: var, normalize, Cholesky 32x32, logdet, trace. One thread per batch.
__global__ void k_final(const float* Szz, const float* Szw, const float* Sww,
                        float* out) {
  __shared__ float Ch[BATCH][32][32];
  int b = threadIdx.x;
  if (b >= BATCH) return;
  const float* S = Szz + b * 1024;
  float mean = 0.f;
  for (int i = 0; i < 32; i++) mean += S[i * 32 + i];
  mean *= (1.f / 32.f);
  float var = fmaxf(mean, 1e-6f);
  float inv = 1.f / var;
  for (int i = 0; i < 32; i++)
    for (int j = 0; j < 32; j++)
      Ch[b][i][j] = S[i * 32 + j] * inv + (i == j ? 1e-3f : 0.f);
  float logdet = 0.f;
  for (int j = 0; j < 32; j++) {
    float d = Ch[b][j][j];
    for (int p = 0; p < j; p++) d -= Ch[b][j][p] * Ch[b][j][p];
    d = sqrtf(fmaxf(d, 1e-30f));
    Ch[b][j][j] = d;
    logdet += 2.f * logf(d);
    float rd = 1.f / d;
    for (int i = j + 1; i < 32; i++) {
      float s = Ch[b][i][j];
      for (int p = 0; p < j; p++) s -= Ch[b][i][p] * Ch[b][j][p];
      Ch[b][i][j] = s * rd;
    }
  }
  float szw2 = 0.f;
  for (int n = 0; n < 32; n++) { float v = Szw[b * 32 + n]; szw2 += v * v; }
  float trace = 1e-3f * Sww[b] + szw2 * inv;
  out[b] = 0.5f * (logdet - trace);
}

// ---------------------------------------------------------------------------
extern "C" void kernel_launch(void* const* d_in, const int* in_sizes, int n_in,
                              void* d_out, int out_size, void* d_ws,
                              size_t ws_size, hipStream_t stream) {
  const float* z[5];
  const float* w[5];
  for (int i = 0; i < 5; i++) {
    z[i] = (const float*)d_in[i];
    w[i] = (const float*)d_in[5 + i];
  }
  const float* L   = (const float*)d_in[10];
  const float* lam = (const float*)d_in[11];

  float* ws  = (float*)d_ws;
  float* Bre = ws + OFF_BRE;
  float* Bim = ws + OFF_BIM;
  float* Vd  = ws + OFF_VD;
  float* Sww = ws + OFF_SWW;
  float* LaT = ws + OFF_LAT;
  float* Lam = ws + OFF_LAM;
  float* Wf  = ws + OFF_WF;
  float* Szz = ws + OFF_SZZ;
  float* Szw = ws + OFF_SZW;

  k_zero<<<2048, 256, 0, stream>>>(ws, (unsigned)ZERO_CNT);
  k_lam<<<(MBASE * 1024 + 255) / 256, 256, 0, stream>>>(lam, Lam);
  k_lat<<<(MBASE * CTP * CTP + 255) / 256, 256, 0, stream>>>(L, LaT);
  k_zfft<<<(CT * NCOL + 255) / 256, 256, 0, stream>>>(z[0], z[1], z[2], z[3],
                                                      z[4], Bre, Bim);
  dim3 gg(NCOL / 64, 3, MBASE);
  k_gemm<<<gg, 192, 0, stream>>>(LaT, Bre, Bim, Lam, Vd);
  k_wfft<<<(BATCH * TPB + 255) / 256, 256, 0, stream>>>(w[0], w[1], w[2], w[3],
                                                        w[4], Wf);
  k_gram<<<BATCH, 192, 0, stream>>>(Vd, Wf, Szz, Szw);
  for (int i = 0; i < 5; i++) {
    int n = in_sizes[5 + i];
    k_ww<<<(n + 255) / 256, 256, 0, stream>>>(w[i], n, n / BATCH, Sww);
  }
  k_final<<<1, 32, 0, stream>>>(Szz, Szw, Sww, (float*)d_out);
}